// QuantumRegressionModel_65481071402129
// MI455X (gfx1250) — compile-verified
//
#include <hip/hip_runtime.h>

typedef __attribute__((ext_vector_type(2))) float v2f;
typedef __attribute__((ext_vector_type(8))) float v8f;

#define NQ     12
#define NSTATE 4096
#define LAYERS 8
#define TPB    256

struct cplx { float r, i; };
__device__ __forceinline__ cplx cmul(cplx a, cplx b) {
    return cplx{ a.r * b.r - a.i * b.i, a.r * b.i + a.i * b.r };
}

__global__ __launch_bounds__(TPB)
void qsim_wmma_kernel(const float* __restrict__ sb,    // (B,8)
                      const float* __restrict__ wts,   // (L,12,3)
                      const float* __restrict__ hw,    // (1,12)
                      const float* __restrict__ hb,    // (1,)
                      float* __restrict__ out)         // (B,)
{
    __shared__ float2 st[NSTATE];              // 32 KB state, resident all circuit
    __shared__ float  wsh[LAYERS * NQ * 3];
    __shared__ float  red;

    const int tid  = threadIdx.x;
    const int b    = blockIdx.x;
    const int lane = tid & 31;
    const int wv   = tid >> 5;       // wave 0..7
    const int col  = lane & 15;      // A row m / B,D column n
    const int half = lane >> 4;      // K half within each 4-wide chunk

    for (int i = tid; i < LAYERS * NQ * 3; i += TPB) wsh[i] = wts[i];

    // |0..0> then RX(sb0)@wire0, RY(sb1)@wire1 -> only 4 nonzero amplitudes
    for (int j = 0; j < NSTATE / TPB; ++j) st[j * TPB + tid] = make_float2(0.f, 0.f);
    if (tid == 0) {
        float s0, c0, s1, c1;
        __sincosf(sb[b * 8 + 0] * 0.5f, &s0, &c0);
        __sincosf(sb[b * 8 + 1] * 0.5f, &s1, &c1);
        st[0]        = make_float2(c0 * c1, 0.f);
        st[1u << 10] = make_float2(c0 * s1, 0.f);
        st[1u << 11] = make_float2(0.f, -s0 * c1);
        st[3u << 10] = make_float2(0.f, -s0 * s1);
        red = 0.f;
    }
    __syncthreads();

    for (int l = 0; l < LAYERS; ++l) {
        // ---- 4 fused 3-qubit gates: Rot(3g) (x) Rot(3g+1) (x) Rot(3g+2) ----
        for (int g = 0; g < 4; ++g) {
            // Per-wire 2x2 complex Rot matrices (uniform across block)
            cplx u[3][4];
            #pragma unroll
            for (int k = 0; k < 3; ++k) {
                const float* gp = &wsh[(l * NQ + 3 * g + k) * 3];
                float phi = gp[0], th = gp[1], om = gp[2];
                float c, s, sa, ca, sbb, cbb;
                __sincosf(th * 0.5f, &s, &c);
                __sincosf((phi + om) * 0.5f, &sa, &ca);
                __sincosf((phi - om) * 0.5f, &sbb, &cbb);
                u[k][0] = cplx{  ca * c,  -sa * c };   // u00 = e^{-ia} c
                u[k][1] = cplx{ -cbb * s, -sbb * s };  // u01 = -e^{ib} s
                u[k][2] = cplx{  cbb * s, -sbb * s };  // u10 = e^{-ib} s
                u[k][3] = cplx{  ca * c,   sa * c };   // u11 = e^{ia} c
            }
            // Lane's row r of U8 = kron(u0,u1,u2); A[m][4c+2*half+v]
            int  m  = col, r = m >> 1;
            int  r2 = r >> 2, r1 = (r >> 1) & 1, r0 = r & 1;
            cplx pa0 = r2 ? u[0][2] : u[0][0], pa1 = r2 ? u[0][3] : u[0][1];
            cplx pb0 = r1 ? u[1][2] : u[1][0], pb1 = r1 ? u[1][3] : u[1][1];
            cplx pc0 = r0 ? u[2][2] : u[2][0], pc1 = r0 ? u[2][3] : u[2][1];
            cplx pcH = half ? pc1 : pc0;
            cplx pab[4] = { cmul(pa0, pb0), cmul(pa0, pb1),
                            cmul(pa1, pb0), cmul(pa1, pb1) };
            bool odd = (m & 1) != 0;
            v2f a[4];
            #pragma unroll
            for (int c = 0; c < 4; ++c) {          // U8[r][q], q = 2c+half
                cplx uq = cmul(pab[c], pcH);
                a[c].x = odd ? uq.i :  uq.r;       // real 2x2 block of cplx mult
                a[c].y = odd ? uq.r : -uq.i;
            }

            const int      sh      = 9 - 3 * g;    // LSB bit position of the oct
            const unsigned lowmask = (1u << sh) - 1u;

            // 512 octs / 8 waves / 16 per chain = 4 chains; batch ALL loads first
            unsigned ibase[4];
            v2f      bfr[4][4];
            #pragma unroll
            for (int ch = 0; ch < 4; ++ch) {
                unsigned o  = (unsigned)(wv * 64 + ch * 16 + col);
                unsigned ib = ((o & ~lowmask) << 3) | (o & lowmask);
                ibase[ch] = ib;
                #pragma unroll
                for (int c = 0; c < 4; ++c) {      // amp q = 2c+half of the oct
                    float2 bv = st[ib + ((unsigned)(2 * c + half) << sh)];
                    bfr[ch][c].x = bv.x; bfr[ch][c].y = bv.y;
                }
            }
            // Interleave two independent accumulation chains to hide the
            // WMMA D->C accumulate latency (two disjoint accumulator blocks).
            #pragma unroll
            for (int cp = 0; cp < 2; ++cp) {
                v8f acc0 = {}, acc1 = {};
                #pragma unroll
                for (int c = 0; c < 4; ++c) {      // K=16 via 4 chained K=4 WMMAs
                    acc0 = __builtin_amdgcn_wmma_f32_16x16x4_f32(
                        false, a[c], false, bfr[2 * cp + 0][c], (short)0, acc0,
                        false, false);
                    acc1 = __builtin_amdgcn_wmma_f32_16x16x4_f32(
                        false, a[c], false, bfr[2 * cp + 1][c], (short)0, acc1,
                        false, false);
                }
                #pragma unroll
                for (int j = 0; j < 4; ++j) {      // D rows 8*half..8*half+7
                    st[ibase[2 * cp + 0] + ((unsigned)(j + 4 * half) << sh)] =
                        make_float2(acc0[2 * j], acc0[2 * j + 1]);
                    st[ibase[2 * cp + 1] + ((unsigned)(j + 4 * half) << sh)] =
                        make_float2(acc1[2 * j], acc1[2 * j + 1]);
                }
            }
            __syncthreads();
        }

        // ---- whole CNOT chain == one permutation: new[y] = old[y ^ (y>>1)] ----
        float2 tmp[NSTATE / TPB];
        #pragma unroll
        for (int j = 0; j < NSTATE / TPB; ++j) {
            unsigned y = (unsigned)(j * TPB + tid);
            tmp[j] = st[y ^ (y >> 1)];
        }
        __syncthreads();
        #pragma unroll
        for (int j = 0; j < NSTATE / TPB; ++j)
            st[(unsigned)(j * TPB + tid)] = tmp[j];
        __syncthreads();
    }

    // out = sum_i |a_i|^2 * (sum_w sign_w(i) * hw[w]) + hb
    float hwv[NQ];
    #pragma unroll
    for (int w = 0; w < NQ; ++w) hwv[w] = hw[w];
    float acc = 0.f;
    for (int j = 0; j < NSTATE / TPB; ++j) {
        unsigned i = (unsigned)(j * TPB + tid);
        float2 amp = st[i];
        float pr = amp.x * amp.x + amp.y * amp.y;
        float coef = 0.f;
        #pragma unroll
        for (int w = 0; w < NQ; ++w)
            coef += ((i >> (11 - w)) & 1u) ? -hwv[w] : hwv[w];
        acc += pr * coef;
    }
    #pragma unroll
    for (int off = 16; off > 0; off >>= 1)
        acc += __shfl_down(acc, off, 32);
    if (lane == 0) atomicAdd(&red, acc);
    __syncthreads();
    if (tid == 0) out[b] = red + hb[0];
}

extern "C" void kernel_launch(void* const* d_in, const int* in_sizes, int n_in,
                              void* d_out, int out_size, void* d_ws, size_t ws_size,
                              hipStream_t stream) {
    (void)n_in; (void)d_ws; (void)ws_size; (void)out_size;
    const float* sb  = (const float*)d_in[0];
    const float* wts = (const float*)d_in[1];
    const float* hw  = (const float*)d_in[2];
    const float* hb  = (const float*)d_in[3];
    float* out = (float*)d_out;
    const int B = in_sizes[0] / 8;   // 8192
    qsim_wmma_kernel<<<B, TPB, 0, stream>>>(sb, wts, hw, hb, out);
}